// APPNPConv_PNY_21474836480042
// MI455X (gfx1250) — compile-verified
//
#include <hip/hip_runtime.h>
#include <math.h>

#define DFEAT 64
#define ALPHA 0.1f
#define EPSN  1e-6f

typedef __attribute__((ext_vector_type(2))) float v2f;
typedef __attribute__((ext_vector_type(8))) float v8f;

// ---------------- setup kernels ----------------

__global__ void zero3_kernel(unsigned* a, unsigned* b, unsigned* c, int n) {
  int i = blockIdx.x * blockDim.x + threadIdx.x;
  if (i < n) { a[i] = 0u; b[i] = 0u; c[i] = 0u; }
}

__global__ void degree_kernel(const int* __restrict__ src, const int* __restrict__ dst,
                              unsigned* out_deg, unsigned* in_deg, int e) {
  int i = blockIdx.x * blockDim.x + threadIdx.x;
  if (i < e) {
    atomicAdd(&out_deg[src[i]], 1u);
    atomicAdd(&in_deg[dst[i]], 1u);
  }
}

__global__ void norm_kernel(const unsigned* __restrict__ out_deg,
                            const unsigned* __restrict__ in_deg,
                            float* src_norm, float* dst_norm, int n) {
  int i = blockIdx.x * blockDim.x + threadIdx.x;
  if (i < n) {
    float od = fmaxf((float)out_deg[i], 1.0f);
    float id = fmaxf((float)in_deg[i], 1.0f);
    src_norm[i] = 1.0f / sqrtf(od);
    dst_norm[i] = 1.0f / sqrtf(id);
  }
}

// Exclusive prefix scan of in-degrees -> row_ptr (two-level, N <= 1024*256)
__global__ void scan_block_kernel(const unsigned* __restrict__ deg,
                                  unsigned* __restrict__ excl,
                                  unsigned* __restrict__ partials, int n) {
  __shared__ unsigned s[256];
  int tid = threadIdx.x;
  int i = blockIdx.x * 256 + tid;
  unsigned v = (i < n) ? deg[i] : 0u;
  s[tid] = v;
  __syncthreads();
  for (int off = 1; off < 256; off <<= 1) {
    unsigned t = (tid >= off) ? s[tid - off] : 0u;
    __syncthreads();
    s[tid] += t;
    __syncthreads();
  }
  if (i < n) excl[i] = s[tid] - v;
  if (tid == 255) partials[blockIdx.x] = s[255];
}

__global__ void scan_partials_kernel(unsigned* partials, int nb) {
  __shared__ unsigned s[1024];
  int tid = threadIdx.x;
  unsigned v = (tid < nb) ? partials[tid] : 0u;
  s[tid] = v;
  __syncthreads();
  for (int off = 1; off < 1024; off <<= 1) {
    unsigned t = (tid >= off) ? s[tid - off] : 0u;
    __syncthreads();
    s[tid] += t;
    __syncthreads();
  }
  if (tid < nb) partials[tid] = s[tid] - v;  // exclusive
}

__global__ void scan_add_kernel(unsigned* excl, const unsigned* __restrict__ partials, int n) {
  int i = blockIdx.x * blockDim.x + threadIdx.x;
  if (i < n) excl[i] += partials[i >> 8];
}

__global__ void csr_fill_kernel(const int* __restrict__ src, const int* __restrict__ dst,
                                const unsigned* __restrict__ row_ptr, unsigned* cursor,
                                unsigned* __restrict__ csr_src, int e) {
  int i = blockIdx.x * blockDim.x + threadIdx.x;
  if (i < e) {
    int d = dst[i];
    unsigned pos = row_ptr[d] + atomicAdd(&cursor[d], 1u);
    csr_src[pos] = (unsigned)src[i];
  }
}

// ---------------- per-iteration kernels ----------------

// One wave (32 lanes) per destination row; lane handles a float2 of the 64-wide
// feature row. Gathers from cur (L2-resident), no atomics, prefetch next row.
__global__ void aggregate_kernel(const float* __restrict__ cur,
                                 const float* __restrict__ src_norm,
                                 const unsigned* __restrict__ row_ptr,
                                 const unsigned* __restrict__ in_deg,
                                 const unsigned* __restrict__ csr_src,
                                 float* __restrict__ agg, int n) {
  int row = blockIdx.x * (blockDim.x >> 5) + (threadIdx.x >> 5);
  if (row >= n) return;
  int lane = threadIdx.x & 31;
  unsigned start = row_ptr[row];
  unsigned len = in_deg[row];
  float ax = 0.0f, ay = 0.0f;
  for (unsigned j = 0; j < len; ++j) {
    unsigned s = csr_src[start + j];
    if (j + 1u < len) {
      unsigned s2 = csr_src[start + j + 1u];
      __builtin_prefetch(cur + (size_t)s2 * DFEAT, 0, 3);  // global_prefetch_b8
    }
    float w = src_norm[s];
    const float2* rp = (const float2*)(cur + (size_t)s * DFEAT);
    float2 v = rp[lane];
    ax = fmaf(v.x, w, ax);
    ay = fmaf(v.y, w, ay);
  }
  float2 o; o.x = ax; o.y = ay;
  ((float2*)(agg + (size_t)row * DFEAT))[lane] = o;
}

// Squared L2 row norms via WMMA: each wave owns 16 rows x 64 features.
// C = A * A^T accumulated over K=64 in 16 chunks of 4 with
// V_WMMA_F32_16X16X4_F32; diagonal of C = squared norms.
// f32 16x4 A layout: lanes 0-15 hold (K=0,1), lanes 16-31 hold (K=2,3).
// f32 4x16 B layout: V0 = rows K0 (lanes 0-15) / K1 (lanes 16-31), V1 = K2/K3.
// f32 16x16 C layout: VGPR r: lanes 0-15 -> (M=r, N=lane); lanes 16-31 -> (M=8+r, N=lane-16).
__global__ void rownorm2_wmma_kernel(const float* __restrict__ mat,
                                     float* __restrict__ out, int n) {
  int wave = blockIdx.x * (blockDim.x >> 5) + (threadIdx.x >> 5);
  int lane = threadIdx.x & 31;
  int base = wave * 16;
  if (base >= n) return;                  // uniform per-wave exit, EXEC stays full otherwise
  int half = lane >> 4;                   // 0: lanes 0-15, 1: lanes 16-31
  int lr   = lane & 15;
  int row  = base + lr;
  if (row >= n) row = n - 1;              // clamp padding lanes (writes are guarded)
  const float* rp = mat + (size_t)row * DFEAT;

  v8f c = {};
#pragma unroll
  for (int kk = 0; kk < 16; ++kk) {
    int k0 = kk * 4;
    v2f a, b;
    a.x = rp[k0 + 2 * half];              // A[row][k0+2h], A[row][k0+2h+1]
    a.y = rp[k0 + 2 * half + 1];
    b.x = rp[k0 + half];                  // B[k0+h][row] = A[row][k0+h]
    b.y = rp[k0 + 2 + half];              // B[k0+2+h][row]
    c = __builtin_amdgcn_wmma_f32_16x16x4_f32(
        /*neg_a=*/false, a, /*neg_b=*/false, b,
        /*c_mod=*/(short)0, c, /*reuse_a=*/false, /*reuse_b=*/false);
  }

  // Extract diagonal: (r,r) lives in VGPR r at lane r; (8+r,8+r) in VGPR r at lane 24+r.
#pragma unroll
  for (int r = 0; r < 8; ++r) {
    int m0 = base + r, m1 = base + 8 + r;
    if (lane == r && m0 < n)      out[m0] = c[r];
    if (lane == 24 + r && m1 < n) out[m1] = c[r];
  }
}

// feat_next = (1-ALPHA) * agg * (||prev||/(||agg||+eps)) * dst_norm + ALPHA * feat0
// where ||prev|| = src_norm[i] * sqrt(pn2[i]).  float4 per thread.
__global__ void combine_kernel(const float* __restrict__ agg,
                               const float* __restrict__ feat0,
                               const float* __restrict__ pn2,
                               const float* __restrict__ an2,
                               const float* __restrict__ src_norm,
                               const float* __restrict__ dst_norm,
                               float* __restrict__ outp, int n) {
  int g = blockIdx.x * blockDim.x + threadIdx.x;   // over n * (DFEAT/4) float4 groups
  int total = n * (DFEAT / 4);
  if (g >= total) return;
  int row = g / (DFEAT / 4);
  float pn = src_norm[row] * sqrtf(pn2[row]);
  float an = sqrtf(an2[row]);
  float scale = (1.0f - ALPHA) * (pn / (an + EPSN)) * dst_norm[row];
  float4 a = ((const float4*)agg)[g];
  float4 f = ((const float4*)feat0)[g];
  float4 o;
  o.x = fmaf(a.x, scale, ALPHA * f.x);
  o.y = fmaf(a.y, scale, ALPHA * f.y);
  o.z = fmaf(a.z, scale, ALPHA * f.z);
  o.w = fmaf(a.w, scale, ALPHA * f.w);
  ((float4*)outp)[g] = o;
}

// ---------------- launch ----------------

extern "C" void kernel_launch(void* const* d_in, const int* in_sizes, int n_in,
                              void* d_out, int out_size, void* d_ws, size_t ws_size,
                              hipStream_t stream) {
  const float* feat0 = (const float*)d_in[0];
  const int*   src   = (const int*)d_in[1];
  const int*   dst   = (const int*)d_in[2];
  int n = in_sizes[0] / DFEAT;   // 100000
  int e = in_sizes[1];           // 3200000
  (void)n_in; (void)out_size; (void)ws_size;

  char* ws = (char*)d_ws;
  size_t off = 0;
  auto alloc = [&](size_t bytes) -> void* {
    void* p = ws + off;
    off = (off + bytes + 255) & ~(size_t)255;
    return p;
  };
  unsigned* out_deg  = (unsigned*)alloc((size_t)n * 4);
  unsigned* in_deg   = (unsigned*)alloc((size_t)n * 4);
  unsigned* cursor   = (unsigned*)alloc((size_t)n * 4);
  unsigned* row_ptr  = (unsigned*)alloc((size_t)n * 4);
  unsigned* partials = (unsigned*)alloc(1024 * 4);
  float*    src_norm = (float*)alloc((size_t)n * 4);
  float*    dst_norm = (float*)alloc((size_t)n * 4);
  float*    pn2      = (float*)alloc((size_t)n * 4);
  float*    an2      = (float*)alloc((size_t)n * 4);
  unsigned* csr_src  = (unsigned*)alloc((size_t)e * 4);
  float*    featA    = (float*)alloc((size_t)n * DFEAT * 4);
  float*    agg      = (float*)alloc((size_t)n * DFEAT * 4);
  float*    outp     = (float*)d_out;

  int nb = (n + 255) / 256;  // 391 <= 1024

  // Degrees, norms, CSR-by-dst (once per launch, deterministic work)
  zero3_kernel<<<(n + 255) / 256, 256, 0, stream>>>(out_deg, in_deg, cursor, n);
  degree_kernel<<<(e + 255) / 256, 256, 0, stream>>>(src, dst, out_deg, in_deg, e);
  norm_kernel<<<(n + 255) / 256, 256, 0, stream>>>(out_deg, in_deg, src_norm, dst_norm, n);
  scan_block_kernel<<<nb, 256, 0, stream>>>(in_deg, row_ptr, partials, n);
  scan_partials_kernel<<<1, 1024, 0, stream>>>(partials, nb);
  scan_add_kernel<<<nb, 256, 0, stream>>>(row_ptr, partials, n);
  csr_fill_kernel<<<(e + 255) / 256, 256, 0, stream>>>(src, dst, row_ptr, cursor, csr_src, e);

  // K=3 propagation steps; ping-pong d_in -> d_out -> featA -> d_out
  const float* cur = feat0;
  float* nexts[3] = { outp, featA, outp };
  for (int k = 0; k < 3; ++k) {
    float* next = nexts[k];
    aggregate_kernel<<<(n + 7) / 8, 256, 0, stream>>>(cur, src_norm, row_ptr, in_deg,
                                                      csr_src, agg, n);
    rownorm2_wmma_kernel<<<(n + 127) / 128, 256, 0, stream>>>(cur, pn2, n);
    rownorm2_wmma_kernel<<<(n + 127) / 128, 256, 0, stream>>>(agg, an2, n);
    combine_kernel<<<((n * (DFEAT / 4)) + 255) / 256, 256, 0, stream>>>(
        agg, feat0, pn2, an2, src_norm, dst_norm, next, n);
    cur = next;
  }
}